// RoIAlign_3882650435973
// MI455X (gfx1250) — compile-verified
//
#include <hip/hip_runtime.h>

// RoI Align, fp32, MI455X (gfx1250).
//
// Roofline: 2000 ROIs x 49 samples x 256 ch ~= 25.1M outputs x ~10 flops
// = 0.25 GFLOP, vs >=264 MB compulsory HBM traffic and ~2-3 GB of L2-grain
// gather traffic -> purely memory bound; WMMA is deliberately absent
// (bilinear sampling has no shared-operand contraction; 16x16xK matrix ops
// would waste >=15/16 of their MACs on a problem that is ~100x memory-bound).
//
// MI455X-specific levers used instead:
//  * 192 MB L2: the 164 MB feature tensor is L2-resident. The 100 MB output
//    stream is write-once/never-read, so it is stored NON-TEMPORAL
//    (__builtin_nontemporal_store -> global_store TH=NT) to avoid evicting
//    the feature map from L2 (164+100 MB would not fit otherwise).
//  * global_prefetch_b8 (gfx1250) across the 800 KB channel stride, which no
//    hardware streamer can cover.
//  * wave32 gathers, 8 waves/block x 2000 blocks for latency hiding.

namespace {

constexpr int   kAH = 7, kAW = 7;
constexpr int   kNS = kAH * kAW;          // 49 samples per ROI
constexpr int   kC  = 256, kH = 200, kW = 200;
constexpr int   kHW = kH * kW;            // 40000 elems = 160 KB channel stride
constexpr float kScale = 0.25f;
constexpr int   kCG    = 5;               // concurrent channel streams per block
constexpr int   kBlock = kNS * kCG;       // 245 threads = 8 wave32, 11 idle lanes

__global__ __launch_bounds__(kBlock) void roi_align_f32(
    const float* __restrict__ feat,   // (B, 256, 200, 200)
    const float* __restrict__ rois,   // (N, 5) = {b, x1, y1, x2, y2}
    float* __restrict__ out,          // (N, 256, 7, 7)
    int n_rois)
{
  const int n = blockIdx.x;
  if (n >= n_rois) return;

  const int t  = threadIdx.x;
  const int s  = t % kNS;               // sample id: s = ph*7 + pw
  const int cg = t / kNS;               // channel stream 0..4
  const int ph = s / kAW;
  const int pw = s % kAW;

  // ROI record: n is uniform across the block -> scalar (SMEM) loads.
  const float* r = rois + (size_t)n * 5;
  const int   b  = (int)r[0];
  const float xs = r[1] * kScale, ys = r[2] * kScale;
  const float xe = r[3] * kScale, ye = r[4] * kScale;

  // Per-sample geometry, computed once per thread (registers only).
  const float in_y = ys + (float)ph * ((ye - ys) / (float)(kAH - 1));
  const float in_x = xs + (float)pw * ((xe - xs) / (float)(kAW - 1));
  const bool valid = (in_y >= 0.0f) & (in_y <= (float)(kH - 1)) &
                     (in_x >= 0.0f) & (in_x <= (float)(kW - 1));
  const float yc  = fminf(fmaxf(in_y, 0.0f), (float)(kH - 1));
  const float xc  = fminf(fmaxf(in_x, 0.0f), (float)(kW - 1));
  const float y0f = floorf(yc), x0f = floorf(xc);
  const int   y0  = (int)y0f,   x0  = (int)x0f;
  const int   y1  = (int)ceilf(yc), x1 = (int)ceilf(xc);
  const float ly  = yc - y0f,   lx  = xc - x0f;

  // 32-bit element offsets (max 4*256*40000 < 2^31).
  const int base = b * (kC * kHW);
  const float* row0 = feat + base + y0 * kW;   // + c*kHW per channel
  const float* row1 = feat + base + y1 * kW;
  float* op = out + (size_t)n * (size_t)(kC * kNS) + s;

  #pragma unroll 2
  for (int c = cg; c < kC; c += kCG) {
    const float* r0 = row0 + c * kHW;
    const float* r1 = row1 + c * kHW;

    // gfx1250 speculative prefetch of this stream's next channel (800 KB
    // stride; lines are L2-resident -> pulls them into the WGP-level cache).
    if (c + kCG < kC) {
      __builtin_prefetch(r0 + kCG * kHW, 0, 1);
      __builtin_prefetch(r1 + kCG * kHW, 0, 1);
    }

    const float tl = r0[x0], tr = r0[x1];
    const float bl = r1[x0], br = r1[x1];

    const float top = tl + (tr - tl) * lx;
    const float bot = bl + (br - bl) * lx;
    float v = top + (bot - top) * ly;
    v = valid ? v : 0.0f;                 // where(my & mx, val, 0)

    // Non-temporal store: keep the 164 MB feature tensor resident in the
    // 192 MB L2 by not caching the 100 MB write-once output stream.
    __builtin_nontemporal_store(v, op + (size_t)c * kNS);
  }
}

} // namespace

extern "C" void kernel_launch(void* const* d_in, const int* in_sizes, int n_in,
                              void* d_out, int out_size, void* d_ws, size_t ws_size,
                              hipStream_t stream) {
  const float* feat = (const float*)d_in[0];   // (4, 256, 200, 200) fp32
  const float* rois = (const float*)d_in[1];   // (N, 5) fp32
  float* out = (float*)d_out;                  // (N, 256, 7, 7) fp32

  const int n_rois = in_sizes[1] / 5;
  if (n_rois <= 0) return;

  hipLaunchKernelGGL(roi_align_f32, dim3(n_rois), dim3(kBlock), 0, stream,
                     feat, rois, out, n_rois);
}